// DecodePredictions_23682449670878
// MI455X (gfx1250) — compile-verified
//
#include <hip/hip_runtime.h>
#include <stdint.h>

// ---------------------------------------------------------------------------
// YOLOv8 decode + greedy NMS for MI455X (gfx1250, wave32).
//
// Stage 1 (decode): memory-bound, ~310 MB in @ 23.3 TB/s => ~13 us floor.
//   64-anchor tiles staged global->LDS with CDNA5 async-to-LDS B128 copies
//   (ASYNCcnt path). LDS layout: row-major per anchor, padded to 148 floats
//   (592 B: 16B-aligned rows for b128; 148 mod 64 = 20 -> only 2-way bank
//   conflict). Compute phase reads 36x ds_load_b128 per anchor.
// Stage 2 (NMS): latency-bound, 100 serial steps/image. One 1024-thread
//   block per image; each thread holds 9 (score, box) pairs in VGPRs.
//   Per step: unrolled local argmax -> wave32 shfl_xor reduce -> 32-wave LDS
//   reduce -> broadcast winner -> register-resident IoU suppression.
// ---------------------------------------------------------------------------

#define N_ANCH   8400
#define NCH      144
#define NCH4     36          // float4s per anchor row
#define PADF     148         // padded row stride in floats (592 B)
#define REG      64
#define NCLS     80
#define BINS     16
#define MAX_OUT  100
#define SCORE_TH 0.3f
#define IOU_TH   0.5f
#define NEG_INF_F (-1e30f)

#define TILE   64
#define NTILES ((N_ANCH + TILE - 1) / TILE)   // 132

// CDNA5 async global->LDS path (ASYNCcnt). Prefer B128, fall back to B32
// (verified working this toolchain), then to plain LDS stores.
#if defined(__HIP_DEVICE_COMPILE__) && \
    __has_builtin(__builtin_amdgcn_global_load_async_to_lds_b128) && \
    __has_builtin(__builtin_amdgcn_s_wait_asynccnt)
#define USE_ASYNC_B128 1
#elif defined(__HIP_DEVICE_COMPILE__) && \
    __has_builtin(__builtin_amdgcn_global_load_async_to_lds_b32) && \
    __has_builtin(__builtin_amdgcn_s_wait_asynccnt)
#define USE_ASYNC_B32 1
#endif

typedef int v4i_t __attribute__((vector_size(16)));
typedef __attribute__((address_space(1))) int*   as1_int_ptr;
typedef __attribute__((address_space(3))) int*   as3_int_ptr;
typedef __attribute__((address_space(1))) v4i_t* as1_v4i_ptr;
typedef __attribute__((address_space(3))) v4i_t* as3_v4i_ptr;

__global__ __launch_bounds__(TILE) void decode_kernel(
    const float* __restrict__ preds, const float* __restrict__ anchors,
    float4* __restrict__ boxes, float* __restrict__ scores,
    int* __restrict__ labels)
{
  __shared__ __align__(16) float lds[PADF * TILE];   // row-major, padded rows

  int blk  = blockIdx.x;
  int b    = blk / NTILES;
  int tile = blk - b * NTILES;
  int a0   = tile * TILE;
  int nvalid = N_ANCH - a0; if (nvalid > TILE) nvalid = TILE;

  const float*  gsrc  = preds + ((size_t)b * N_ANCH + a0) * NCH;
  const float4* gsrc4 = (const float4*)gsrc;        // 16B-aligned: 576|b,a0 offsets
  int nf4 = nvalid * NCH4;

  // Coalesced staging: chunk g = consecutive float4s across threads.
  // Row r = g/36, col c4 = g%36 -> LDS float offset r*148 + c4*4 (16B aligned).
  for (int g = threadIdx.x; g < nf4; g += TILE) {
    int r  = g / NCH4;
    int c4 = g - r * NCH4;
    float* dst = lds + r * PADF + c4 * 4;
#if defined(USE_ASYNC_B128)
    __builtin_amdgcn_global_load_async_to_lds_b128(
        (as1_v4i_ptr)(uintptr_t)(gsrc4 + g),
        (as3_v4i_ptr)(uint32_t)(uintptr_t)dst, 0, 0);
#elif defined(USE_ASYNC_B32)
    const float* src = (const float*)(gsrc4 + g);
#pragma unroll
    for (int u = 0; u < 4; ++u) {
      __builtin_amdgcn_global_load_async_to_lds_b32(
          (as1_int_ptr)(uintptr_t)(src + u),
          (as3_int_ptr)(uint32_t)(uintptr_t)(dst + u), 0, 0);
    }
#else
    *(float4*)dst = gsrc4[g];
#endif
  }
#if defined(USE_ASYNC_B128) || defined(USE_ASYNC_B32)
  __builtin_amdgcn_s_wait_asynccnt(0);
#endif
  __syncthreads();

  int a = threadIdx.x;
  if (a < nvalid) {
    int n = a0 + a;
    const float4* row = (const float4*)(lds + a * PADF);

    // DFL decode: 4x softmax(16) dot arange(16); rows read as float4s.
    float d[4];
#pragma unroll
    for (int k = 0; k < 4; ++k) {
      float l[BINS];
#pragma unroll
      for (int q = 0; q < 4; ++q) {
        float4 v = row[k * 4 + q];
        l[q * 4 + 0] = v.x; l[q * 4 + 1] = v.y;
        l[q * 4 + 2] = v.z; l[q * 4 + 3] = v.w;
      }
      float m = -3.402823466e38f;
#pragma unroll
      for (int j = 0; j < BINS; ++j) m = fmaxf(m, l[j]);
      float s = 0.f, w = 0.f;
#pragma unroll
      for (int j = 0; j < BINS; ++j) {
        float e = expf(l[j] - m);
        s += e;
        w += e * (float)j;
      }
      d[k] = w / s;
    }
    float4 an  = ((const float4*)anchors)[n];
    float hw0 = an.z - an.x, hw1 = an.w - an.y;
    float c0  = (an.x + an.z) * 0.5f, c1 = (an.y + an.w) * 0.5f;
    float bc0 = (d[2] - d[0]) * 0.5f * hw0 + c0;
    float bc1 = (d[3] - d[1]) * 0.5f * hw1 + c1;
    float bh  = (d[2] + d[0]) * hw0;
    float bw  = (d[3] + d[1]) * hw1;
    size_t oi = (size_t)b * N_ANCH + n;
    boxes[oi] = make_float4(bc0 - bh * 0.5f, bc1 - bw * 0.5f,
                            bc0 + bh * 0.5f, bc1 + bw * 0.5f);

    // class max / argmax (first-index-wins via strict > ascending scan)
    float best = -3.402823466e38f;
    int   bi   = 0;
#pragma unroll
    for (int q = 0; q < NCLS / 4; ++q) {
      float4 v = row[16 + q];
      int c = q * 4;
      if (v.x > best) { best = v.x; bi = c; }
      if (v.y > best) { best = v.y; bi = c + 1; }
      if (v.z > best) { best = v.z; bi = c + 2; }
      if (v.w > best) { best = v.w; bi = c + 3; }
    }
    scores[oi] = best;
    labels[oi] = bi;
  }
}

// ---------------------------------------------------------------------------
// NMS: one block (1024 threads = 32 wave32) per image.
// ---------------------------------------------------------------------------
#define NMS_T 1024
#define ITEMS ((N_ANCH + NMS_T - 1) / NMS_T)   // 9

__global__ __launch_bounds__(NMS_T) void nms_kernel(
    const float4* __restrict__ boxes, const float* __restrict__ scores,
    const int* __restrict__ labels, float* __restrict__ out, int nb)
{
  int b = blockIdx.x;
  int t = threadIdx.x;
  const size_t base = (size_t)b * N_ANCH;

  float  sc[ITEMS];
  float4 bx[ITEMS];
#pragma unroll
  for (int k = 0; k < ITEMS; ++k) {
    int i = t + k * NMS_T;
    if (i < N_ANCH) {
      float s = scores[base + i];
      sc[k] = (s > SCORE_TH) ? s : NEG_INF_F;
      bx[k] = boxes[base + i];
    } else {
      sc[k] = NEG_INF_F;
      bx[k] = make_float4(0.f, 0.f, 0.f, 0.f);
    }
  }

  __shared__ float red_v[32];
  __shared__ int   red_i[32];
  __shared__ float win[5];   // y1,x1,y2,x2,score of winner
  __shared__ int   widx;

  float* ob   = out + (size_t)b * MAX_OUT * 4;
  float* olab = out + (size_t)nb * MAX_OUT * 4 + (size_t)b * MAX_OUT;
  float* osc  = out + (size_t)nb * MAX_OUT * 5 + (size_t)b * MAX_OUT;

  int lane = t & 31;
  int wave = t >> 5;

  for (int s = 0; s < MAX_OUT; ++s) {
    // ---- argmax (value max, smallest index on ties == jnp.argmax) ----
    float bv = sc[0];
    int   bi = t;
#pragma unroll
    for (int k = 1; k < ITEMS; ++k) {
      float v = sc[k];
      int   i = t + k * NMS_T;
      if (v > bv) { bv = v; bi = i; }
    }
#pragma unroll
    for (int off = 16; off > 0; off >>= 1) {
      float ov = __shfl_xor(bv, off, 32);
      int   oi = __shfl_xor(bi, off, 32);
      if (ov > bv || (ov == bv && oi < bi)) { bv = ov; bi = oi; }
    }
    if (lane == 0) { red_v[wave] = bv; red_i[wave] = bi; }
    __syncthreads();                                   // B1
    if (t < 32) {
      bv = red_v[t]; bi = red_i[t];
#pragma unroll
      for (int off = 16; off > 0; off >>= 1) {
        float ov = __shfl_xor(bv, off, 32);
        int   oi = __shfl_xor(bi, off, 32);
        if (ov > bv || (ov == bv && oi < bi)) { bv = ov; bi = oi; }
      }
      if (t == 0) { win[4] = bv; widx = bi; }
    }
    __syncthreads();                                   // B2

    float wv = win[4];
    int   wj = widx;
    if (!(wv > NEG_INF_F * 0.5f)) {
      // all remaining suppressed forever -> fill tail with invalid entries
      for (int q = s + t; q < MAX_OUT; q += NMS_T) {
        ob[q * 4 + 0] = 0.f; ob[q * 4 + 1] = 0.f;
        ob[q * 4 + 2] = 0.f; ob[q * 4 + 3] = 0.f;
        olab[q] = -1.f; osc[q] = 0.f;
      }
      break;   // uniform across block
    }

    // owner thread publishes winner box from its registers
    if ((wj & (NMS_T - 1)) == t) {
      int kk = wj >> 10;
#pragma unroll
      for (int k = 0; k < ITEMS; ++k) {
        if (k == kk) {
          win[0] = bx[k].x; win[1] = bx[k].y;
          win[2] = bx[k].z; win[3] = bx[k].w;
        }
      }
    }
    __syncthreads();                                   // B3

    float wy1 = win[0], wx1 = win[1], wy2 = win[2], wx2 = win[3];
    if (t == 0) {
      ob[s * 4 + 0] = wy1; ob[s * 4 + 1] = wx1;
      ob[s * 4 + 2] = wy2; ob[s * 4 + 3] = wx2;
      osc[s]  = wv;
      olab[s] = (float)labels[base + wj];
    }

    // ---- register-resident IoU suppression ----
    float a1 = (wy2 - wy1) * (wx2 - wx1);
#pragma unroll
    for (int k = 0; k < ITEMS; ++k) {
      int i = t + k * NMS_T;
      if (i == wj) { sc[k] = NEG_INF_F; continue; }
      if (sc[k] > NEG_INF_F * 0.5f) {
        float y1 = bx[k].x, x1 = bx[k].y, y2 = bx[k].z, x2 = bx[k].w;
        float ty  = fmaxf(wy1, y1), tx = fmaxf(wx1, x1);
        float by  = fminf(wy2, y2), bxx = fminf(wx2, x2);
        float ih  = fmaxf(by - ty, 0.f), iw = fmaxf(bxx - tx, 0.f);
        float inter = ih * iw;
        float a2 = (y2 - y1) * (x2 - x1);
        float iou = inter / (a1 + a2 - inter + 1e-9f);
        if (iou > IOU_TH) sc[k] = NEG_INF_F;
      }
    }
    // hazards on red_*/win/widx across iterations are separated by B1/B2/B3
  }
}

// ---------------------------------------------------------------------------
extern "C" void kernel_launch(void* const* d_in, const int* in_sizes, int n_in,
                              void* d_out, int out_size, void* d_ws, size_t ws_size,
                              hipStream_t stream)
{
  (void)n_in; (void)out_size; (void)ws_size;
  const float* preds   = (const float*)d_in[0];
  const float* anchors = (const float*)d_in[1];
  float* out = (float*)d_out;

  int B = in_sizes[0] / (N_ANCH * NCH);   // 64

  char* ws = (char*)d_ws;
  size_t boxes_bytes = (size_t)B * N_ANCH * 4 * sizeof(float);   // 8.6 MB
  size_t sc_bytes    = (size_t)B * N_ANCH * sizeof(float);       // 2.15 MB
  float4* boxes  = (float4*)ws;
  float*  scores = (float*)(ws + boxes_bytes);
  int*    labels = (int*)(ws + boxes_bytes + sc_bytes);

  decode_kernel<<<B * NTILES, TILE, 0, stream>>>(preds, anchors, boxes, scores, labels);
  nms_kernel<<<B, NMS_T, 0, stream>>>(boxes, scores, labels, out, B);
}